// GCN_22462678958349
// MI455X (gfx1250) — compile-verified
//
#include <hip/hip_runtime.h>
#include <stdint.h>

typedef float v2f __attribute__((ext_vector_type(2)));
typedef float v4f __attribute__((ext_vector_type(4)));
typedef float v8f __attribute__((ext_vector_type(8)));

#define IN_DIM 512
#define HID 16

// ---------------------------------------------------------------------------
// Degree / normalization
// ---------------------------------------------------------------------------
__global__ __launch_bounds__(256) void gcn_init_deg(float* __restrict__ deg, int n) {
  int i = blockIdx.x * 256 + threadIdx.x;
  if (i < n) deg[i] = 1.0f;  // self-loop contributes 1 to every degree
}

__global__ __launch_bounds__(256) void gcn_count_deg(const long long* __restrict__ dst,
                                                     float* __restrict__ deg,
                                                     long long nE) {
  long long i = (long long)blockIdx.x * 256 + threadIdx.x;
  if (i < nE) {
    int d = (int)__builtin_nontemporal_load(&dst[i]);  // streamed, don't pollute L2
    atomicAdd(&deg[d], 1.0f);                          // deg is 400KB, L2-resident
  }
}

__global__ __launch_bounds__(256) void gcn_dinv(float* __restrict__ deg, int n) {
  int i = blockIdx.x * 256 + threadIdx.x;
  if (i < n) {
    float dg = deg[i];
    deg[i] = dg > 0.0f ? rsqrtf(dg) : 0.0f;  // in-place: deg -> dinv
  }
}

// ---------------------------------------------------------------------------
// Layer-1 GEMM: t1 = x @ W1   ([N,512] x [512,16]), f32 WMMA 16x16x4
// One 16-row tile per wave, 8 waves/block. x staged via LDS (NT loads),
// W1 resident in LDS. 128 v_wmma_f32_16x16x4_f32 per tile.
// ---------------------------------------------------------------------------
__global__ __launch_bounds__(256) void gcn_gemm1_wmma(const float* __restrict__ x,
                                                      const float* __restrict__ W1,
                                                      float* __restrict__ t1,
                                                      int nNodes) {
  __shared__ float sW[IN_DIM * HID];  // 32 KB, loaded once
  __shared__ float sX[8][16 * 36];    // 18 KB, stride 36 words -> bank-conflict-free
  const int tid  = threadIdx.x;
  const int wave = tid >> 5;
  const int lane = tid & 31;
  const int half = lane >> 4;   // 0: lanes 0-15, 1: lanes 16-31
  const int lrow = lane & 15;

  for (int i = tid; i < IN_DIM * HID; i += 256) sW[i] = W1[i];
  __syncthreads();

  const int tile    = blockIdx.x * 8 + wave;
  const int rowBase = tile * 16;
  v8f acc = {0.f, 0.f, 0.f, 0.f, 0.f, 0.f, 0.f, 0.f};

  for (int kc = 0; kc < IN_DIM; kc += 32) {
    // Stage this wave's 16x32 fp32 chunk of x (coalesced 128B lines, NT).
    #pragma unroll
    for (int j = 0; j < 4; ++j) {
      int slot = lane + j * 32;        // 0..127 = 16 rows x 8 float4 slots
      int r    = slot >> 3;
      int c4   = (slot & 7) << 2;
      int grow = rowBase + r;
      if (grow >= nNodes) grow = nNodes - 1;  // clamp tail (stores guarded later)
      v4f v = __builtin_nontemporal_load(
          (const v4f*)&x[(size_t)grow * IN_DIM + kc + c4]);
      *(v4f*)&sX[wave][r * 36 + c4] = v;
    }
    __syncthreads();
    #pragma unroll
    for (int k = 0; k < 8; ++k) {
      // A 16x4 f32: lane<16 holds (M=lrow, K=4k..4k+1); lane>=16 K=4k+2..4k+3
      v2f a = *(const v2f*)&sX[wave][lrow * 36 + (k << 2) + (half << 1)];
      // B 4x16 f32: lane<16 holds rows K=4k,4k+1 at N=lrow; lane>=16 rows 4k+2,4k+3
      int kr = kc + (k << 2) + (half << 1);
      v2f b;
      b.x = sW[(kr + 0) * HID + lrow];
      b.y = sW[(kr + 1) * HID + lrow];
      acc = __builtin_amdgcn_wmma_f32_16x16x4_f32(false, a, false, b,
                                                  (short)0, acc, false, false);
    }
    __syncthreads();
  }

  // C/D: VGPR j <-> M = half*8 + j, N = lrow
  const int mbase = rowBase + half * 8;
  #pragma unroll
  for (int j = 0; j < 8; ++j) {
    int row = mbase + j;
    if (row < nNodes) t1[(size_t)row * HID + lrow] = acc[j];
  }
}

// ---------------------------------------------------------------------------
// Layer-2 GEMM fused with bias+ReLU of layer 1:
//   t2 = relu(agg1 + b1) @ W2   ([N,16] x [16,16]), 4 WMMA steps, K=16.
// agg1 is 6.4MB -> L2-resident, direct global float2 loads are fine.
// ---------------------------------------------------------------------------
__global__ __launch_bounds__(256) void gcn_gemm2_wmma(const float* __restrict__ agg1,
                                                      const float* __restrict__ b1,
                                                      const float* __restrict__ W2,
                                                      float* __restrict__ t2,
                                                      int nNodes) {
  const int tid  = threadIdx.x;
  const int wave = tid >> 5;
  const int lane = tid & 31;
  const int half = lane >> 4;
  const int lrow = lane & 15;
  const int tile    = blockIdx.x * 8 + wave;
  const int rowBase = tile * 16;
  int grow = rowBase + lrow;
  if (grow >= nNodes) grow = nNodes - 1;

  v8f acc = {0.f, 0.f, 0.f, 0.f, 0.f, 0.f, 0.f, 0.f};
  #pragma unroll
  for (int k = 0; k < 4; ++k) {
    int col = (k << 2) + (half << 1);
    v2f a = *(const v2f*)&agg1[(size_t)grow * HID + col];
    a.x = fmaxf(a.x + b1[col + 0], 0.0f);   // fuse bias + ReLU into A fragment
    a.y = fmaxf(a.y + b1[col + 1], 0.0f);
    v2f b;
    b.x = W2[(col + 0) * HID + lrow];
    b.y = W2[(col + 1) * HID + lrow];
    acc = __builtin_amdgcn_wmma_f32_16x16x4_f32(false, a, false, b,
                                                (short)0, acc, false, false);
  }

  const int mbase = rowBase + half * 8;
  #pragma unroll
  for (int j = 0; j < 8; ++j) {
    int row = mbase + j;
    if (row < nNodes) t2[(size_t)row * HID + lrow] = acc[j];
  }
}

// ---------------------------------------------------------------------------
// Self-loop init: agg[i,:] = dinv[i]^2 * t[i,:] (+ bias). Fully overwrites agg.
// ---------------------------------------------------------------------------
__global__ __launch_bounds__(256) void gcn_self_init(const float* __restrict__ dinv,
                                                     const float* __restrict__ t,
                                                     const float* __restrict__ bias,
                                                     int useBias,
                                                     float* __restrict__ agg, int n) {
  int gid = blockIdx.x * 256 + threadIdx.x;
  int i = gid >> 2;
  if (i >= n) return;
  int q = gid & 3;
  float di = dinv[i];
  float c  = di * di;
  v4f v = ((const v4f*)t)[(size_t)i * 4 + q];
  v4f o = v * c;
  if (useBias) o += ((const v4f*)bias)[q];
  ((v4f*)agg)[(size_t)i * 4 + q] = o;
}

// ---------------------------------------------------------------------------
// Edge scatter: agg[dst,:] += dinv[src]*dinv[dst] * t[src,:]
// t/agg are 6.4MB (L2-resident); edge list streamed NT; f32 atomics hit L2.
// ---------------------------------------------------------------------------
__global__ __launch_bounds__(256) void gcn_edge_agg(const long long* __restrict__ src,
                                                    const long long* __restrict__ dst,
                                                    const float* __restrict__ dinv,
                                                    const float* __restrict__ t,
                                                    float* __restrict__ agg,
                                                    long long nE) {
  long long e = (long long)blockIdx.x * 256 + threadIdx.x;
  if (e >= nE) return;
  int s = (int)__builtin_nontemporal_load(&src[e]);
  int d = (int)__builtin_nontemporal_load(&dst[e]);
  float c = dinv[s] * dinv[d];
  const v4f* ts = (const v4f*)&t[(size_t)s * HID];
  float*     ad = &agg[(size_t)d * HID];
  #pragma unroll
  for (int q = 0; q < 4; ++q) {
    v4f v = ts[q];
    atomicAdd(&ad[q * 4 + 0], c * v.x);
    atomicAdd(&ad[q * 4 + 1], c * v.y);
    atomicAdd(&ad[q * 4 + 2], c * v.z);
    atomicAdd(&ad[q * 4 + 3], c * v.w);
  }
}

// ---------------------------------------------------------------------------
extern "C" void kernel_launch(void* const* d_in, const int* in_sizes, int n_in,
                              void* d_out, int out_size, void* d_ws, size_t ws_size,
                              hipStream_t stream) {
  (void)n_in; (void)out_size; (void)ws_size;
  const float*     x  = (const float*)d_in[0];
  const long long* ei = (const long long*)d_in[1];
  const float*     W1 = (const float*)d_in[2];
  const float*     b1 = (const float*)d_in[3];
  const float*     W2 = (const float*)d_in[4];
  const float*     b2 = (const float*)d_in[5];
  float* out = (float*)d_out;

  const int       N = in_sizes[0] / IN_DIM;
  const long long E = (long long)in_sizes[1] / 2;
  const long long* srcI = ei;       // edge_index[0]
  const long long* dstI = ei + E;   // edge_index[1]

  // Workspace layout (floats): dinv[N] | t1[N*16] | agg1[N*16] | t2[N*16]
  float* dinv = (float*)d_ws;
  float* t1   = dinv + N;
  float* agg1 = t1 + (size_t)N * HID;
  float* t2   = agg1 + (size_t)N * HID;

  const int nBlkN  = (N + 255) / 256;
  const int nBlkN4 = (N * 4 + 255) / 256;
  const int nBlkE  = (int)((E + 255) / 256);
  const int nTiles = (N + 15) / 16;
  const int nBlkG  = (nTiles + 7) / 8;

  // Normalization coefficients
  gcn_init_deg <<<nBlkN, 256, 0, stream>>>(dinv, N);
  gcn_count_deg<<<nBlkE, 256, 0, stream>>>(dstI, dinv, E);
  gcn_dinv     <<<nBlkN, 256, 0, stream>>>(dinv, N);

  // Layer 1: t1 = x@W1 ; agg1 = D^-1/2 (A+I) D^-1/2 t1
  gcn_gemm1_wmma<<<nBlkG, 256, 0, stream>>>(x, W1, t1, N);
  gcn_self_init <<<nBlkN4, 256, 0, stream>>>(dinv, t1, b1, 0, agg1, N);
  gcn_edge_agg  <<<nBlkE, 256, 0, stream>>>(srcI, dstI, dinv, t1, agg1, E);

  // Layer 2: t2 = relu(agg1+b1)@W2 ; out = D^-1/2 (A+I) D^-1/2 t2 + b2
  gcn_gemm2_wmma<<<nBlkG, 256, 0, stream>>>(agg1, b1, W2, t2, N);
  gcn_self_init <<<nBlkN4, 256, 0, stream>>>(dinv, t2, b2, 1, out, N);
  gcn_edge_agg  <<<nBlkE, 256, 0, stream>>>(srcI, dstI, dinv, t2, out, E);
}